// MambaBlock_47201690583665
// MI455X (gfx1250) — compile-verified
//
#include <hip/hip_runtime.h>
#include <hip/hip_bf16.h>

// ---------------- problem constants (from reference) ----------------
#define DIM     1024
#define DSTATE  16
#define DCONV   4
#define DINNER  2048          // EXPAND * DIM
#define BSZ     4
#define SSZ     2048
#define MROWS   (BSZ * SSZ)   // 8192 token rows

typedef float v2f __attribute__((ext_vector_type(2)));
typedef float v4f __attribute__((ext_vector_type(4)));
typedef float v8f __attribute__((ext_vector_type(8)));
typedef unsigned int u32x4 __attribute__((ext_vector_type(4)));
typedef int i32x4 __attribute__((ext_vector_type(4)));
typedef int i32x8 __attribute__((ext_vector_type(8)));

__device__ __forceinline__ float softplusf(float x) {
    // jax.nn.softplus = log1p(exp(x)) with large-x passthrough
    return (x > 20.0f) ? x : log1pf(expf(x));
}

// =====================================================================
// FP32 WMMA GEMM:  out[m,n] = sum_k A[m,k] * W[n,k] + bias[n]
// A: MxK row-major, W: NxK row-major (i.e. computes A @ W^T + b).
// Block: 256 threads (8 wave32), tile 64(M) x 128(N), K staged via LDS
// in chunks of 32. Each wave computes a 32x32 tile = 2x2 WMMA frags.
//
// Tile staging uses the Tensor Data Mover (TDM): wave 0 issues two 2D
// tensor_load_to_lds descriptors (A 64x32, B 128x32 f32) per K-chunk,
// with TDM LDS padding (pad_interval=4 -> every 32 dwords, pad_amount=3
// -> +4 dwords) producing exactly the 36-float padded row stride used
// for bank-conflict-free WMMA fragment reads. Completion is tracked by
// TENSORcnt; the workgroup barrier publishes the tiles.
//
// V_WMMA_F32_16X16X4_F32 lane layouts (ISA 7.12.2):
//   A(16x4):  lane L holds row (L&15), K = (L>>4)*2 + {0,1}  -> v2f
//   B(4x16):  lane L holds col (L&15), K = (L>>4)*2 + {0,1}  -> v2f
//   C/D:      VGPR r holds M = r + 8*(L>>4), N = (L&15)       -> v8f
// =====================================================================
#define SPAD 36   // 32 + 4 pad: stride 144B = 9*16B -> aligned v4f, no bank conflicts
#define GEMM_LDS_BYTES ((64 + 128) * SPAD * 4)   // 27648 B dynamic LDS
#define SB_LDS_OFF (64 * SPAD * 4)               // 9216: sB starts after sA

#if defined(__HIP_DEVICE_COMPILE__) && __has_builtin(__builtin_amdgcn_tensor_load_to_lds)
#define USE_TDM 1
#else
#define USE_TDM 0
#endif

#if USE_TDM
// Build + issue one 2D TDM tile load: rows x 32 f32 elements, row stride
// kstride elements, into LDS at lds_off with 4-dword row padding.
// D# packing per CDNA5 ISA 8.3 (group0) / 8.4 (group1).
// clang-23 toolchain: 6-arg builtin (g0 u32x4, g1 i32x8, g2 i32x4,
// g3 i32x4, g4 i32x8, cpol i32); groups past 1 are unused for 2D tiles.
__device__ __forceinline__ void tdm_load_tile(unsigned lds_off, const float* gsrc,
                                              unsigned rows, unsigned kstride)
{
    const unsigned long long ga = (unsigned long long)(uintptr_t)gsrc;
    u32x4 g0;
    g0[0] = 1u;                                                 // count=1 (valid), user mode
    g0[1] = lds_off;                                            // lds_addr (bytes)
    g0[2] = (unsigned)(ga & 0xffffffffu);                       // global_addr[31:0]
    g0[3] = (unsigned)((ga >> 32) & 0x01ffffffu) | (2u << 30);  // global_addr[56:32] | type=2

    const unsigned td0 = 0x100000u, td1 = 0x100000u;            // loose OOB bounds (tile in-range)
    i32x8 g1;
    g1[0] = (int)((2u << 16)        // data_size = 4 bytes
                | (1u << 20)        // pad_enable
                | (4u << 22)        // pad_interval: every 32 dwords (one 128B row)
                | (3u << 25));      // pad_amount: 4 dwords -> 36-float row stride
    g1[1] = (int)((td0 & 0xffffu) << 16);                       // abar=0 | tensor_dim0[15:0]
    g1[2] = (int)((td0 >> 16) | ((td1 & 0xffffu) << 16));       // tensor_dim0[31:16] | tensor_dim1[15:0]
    g1[3] = (int)((td1 >> 16) | (32u << 16));                   // tensor_dim1[31:16] | tile_dim0=32
    g1[4] = (int)(rows & 0xffffu);                              // tile_dim1=rows | tile_dim2=0
    g1[5] = (int)kstride;                                       // tensor_dim0_stride[31:0]
    g1[6] = 0;                                                  // stride0[47:32] | stride1[15:0]
    g1[7] = 0;                                                  // stride1[47:16]

    i32x4 gz4 = {0, 0, 0, 0};                                   // 2D tensor: groups 2/3 unused
    i32x8 gz8 = {0, 0, 0, 0, 0, 0, 0, 0};                       // unused trailing group
    __builtin_amdgcn_tensor_load_to_lds(g0, g1, gz4, gz4, gz8, 0);
}
#endif

__global__ __launch_bounds__(256)
void k_gemm_f32(const float* __restrict__ A, const float* __restrict__ W,
                const float* __restrict__ bias,
                float* __restrict__ out0, float* __restrict__ out1,
                int K, int nsplit, int ld0, int ld1)
{
    extern __shared__ float smem[];          // dynamic LDS: base offset 0
    float* sA = smem;                        // [64  x SPAD]
    float* sB = smem + 64 * SPAD;            // [128 x SPAD]

    const int tid  = threadIdx.x;
    const int wave = tid >> 5;
    const int lane = tid & 31;
    const int half = lane >> 4;   // EXEC must be all-ones for WMMA: no divergence here
    const int lq   = lane & 15;
    const int wm   = wave & 1;    // wave M index (0..1)
    const int wn   = wave >> 1;   // wave N index (0..3)
    const int m0   = blockIdx.y * 64;
    const int n0   = blockIdx.x * 128;

    v8f acc[2][2] = {};

    for (int kb = 0; kb < K; kb += 32) {
        __syncthreads();   // previous chunk fully consumed before overwrite
#if USE_TDM
        if (wave == 0) {   // wave-uniform: single TDM issue per tile
            tdm_load_tile(0u,         A + (size_t)m0 * K + kb,  64u, (unsigned)K);
            tdm_load_tile(SB_LDS_OFF, W + (size_t)n0 * K + kb, 128u, (unsigned)K);
            __builtin_amdgcn_s_wait_tensorcnt(0);
        }
#else
#pragma unroll
        for (int it = 0; it < 2; ++it) {
            int idx = it * 256 + tid;
            int row = idx >> 3, c4 = idx & 7;
            *(v4f*)(sA + row * SPAD + c4 * 4) =
                *(const v4f*)(A + (size_t)(m0 + row) * K + kb + c4 * 4);
        }
#pragma unroll
        for (int it = 0; it < 4; ++it) {
            int idx = it * 256 + tid;
            int row = idx >> 3, c4 = idx & 7;
            *(v4f*)(sB + row * SPAD + c4 * 4) =
                *(const v4f*)(W + (size_t)(n0 + row) * K + kb + c4 * 4);
        }
#endif
        __syncthreads();   // publish tiles to all waves

#pragma unroll
        for (int k4 = 0; k4 < 8; ++k4) {
            const int kk = k4 * 4 + half * 2;
            v2f a0 = *(const v2f*)(sA + (wm * 32 +  0 + lq) * SPAD + kk);
            v2f a1 = *(const v2f*)(sA + (wm * 32 + 16 + lq) * SPAD + kk);
            v2f b0 = *(const v2f*)(sB + (wn * 32 +  0 + lq) * SPAD + kk);
            v2f b1 = *(const v2f*)(sB + (wn * 32 + 16 + lq) * SPAD + kk);
            acc[0][0] = __builtin_amdgcn_wmma_f32_16x16x4_f32(false, a0, false, b0, (short)0, acc[0][0], false, false);
            acc[0][1] = __builtin_amdgcn_wmma_f32_16x16x4_f32(false, a0, false, b1, (short)0, acc[0][1], false, false);
            acc[1][0] = __builtin_amdgcn_wmma_f32_16x16x4_f32(false, a1, false, b0, (short)0, acc[1][0], false, false);
            acc[1][1] = __builtin_amdgcn_wmma_f32_16x16x4_f32(false, a1, false, b1, (short)0, acc[1][1], false, false);
        }
    }

    // bias-fused epilogue with output split (half-wave writes 64B runs of n)
#pragma unroll
    for (int nt = 0; nt < 2; ++nt) {
        const int n = n0 + wn * 32 + nt * 16 + lq;
        const float bv = bias[n];
#pragma unroll
        for (int mt = 0; mt < 2; ++mt) {
#pragma unroll
            for (int r = 0; r < 8; ++r) {
                const int m = m0 + wm * 32 + mt * 16 + r + 8 * half;
                const float val = acc[mt][nt][r] + bv;
                if (n < nsplit) out0[(size_t)m * ld0 + n] = val;
                else            out1[(size_t)m * ld1 + (n - nsplit)] = val;
            }
        }
    }
}

// =====================================================================
// Depthwise conv, kernel 4, pad (1,2) along t (cross-correlation, per JAX)
// conv[b,t,d] = cb[d] + sum_k x[b, t-1+k, d] * cw[d,k]
// =====================================================================
__global__ void k_conv(const float* __restrict__ xcopy, const float* __restrict__ cw,
                       const float* __restrict__ cb, float* __restrict__ convs)
{
    const int d = blockIdx.x * blockDim.x + threadIdx.x;
    const int t = blockIdx.y;
    const int b = blockIdx.z;
    const float* xr = xcopy + (size_t)b * SSZ * DINNER + d;
    float acc = cb[d];
    if (t >= 1)       acc = fmaf(xr[(size_t)(t - 1) * DINNER], cw[d * 4 + 0], acc);
                      acc = fmaf(xr[(size_t)(t    ) * DINNER], cw[d * 4 + 1], acc);
    if (t + 1 < SSZ)  acc = fmaf(xr[(size_t)(t + 1) * DINNER], cw[d * 4 + 2], acc);
    if (t + 2 < SSZ)  acc = fmaf(xr[(size_t)(t + 2) * DINNER], cw[d * 4 + 3], acc);
    convs[((size_t)b * SSZ + t) * DINNER + d] = acc;
}

// =====================================================================
// Pack the only live projection rows: dt_w (16 rows) + xproj_w row 20
// (gamma) into a transposed [c][17] matrix for coalesced lane reads.
// =====================================================================
__global__ void k_prep(const float* __restrict__ dt_w, const float* __restrict__ xproj_w,
                       float* __restrict__ wproj)
{
    const int i = blockIdx.x * blockDim.x + threadIdx.x;
    if (i >= DINNER * 17) return;
    const int c = i / 17, j = i % 17;
    wproj[i] = (j < 16) ? dt_w[j * DINNER + c] : xproj_w[20 * DINNER + c];
}

// =====================================================================
// Per token row: 17 dot products over DINNER channels.
// n<16: dA = exp(-softplus(dot+dt_b[n]) * softplus(A[n]))
// n=16: gamma -> silu(gamma) stored directly.
// One wave per row; conv row reads are lane-uniform (broadcast).
// =====================================================================
__global__ __launch_bounds__(32)
void k_delta(const float* __restrict__ convs, const float* __restrict__ wproj,
             const float* __restrict__ dt_b, const float* __restrict__ xproj_b,
             const float* __restrict__ Avec, float* __restrict__ dAbuf,
             float* __restrict__ gsilu)
{
    const int row = blockIdx.x;
    const int n   = threadIdx.x;            // 0..31, 17 active
    const float* crow = convs + (size_t)row * DINNER;
    float acc = 0.0f;
    if (n < 17) {
        for (int c = 0; c < DINNER; c += 4) {
            acc = fmaf(crow[c + 0], wproj[(c + 0) * 17 + n], acc);
            acc = fmaf(crow[c + 1], wproj[(c + 1) * 17 + n], acc);
            acc = fmaf(crow[c + 2], wproj[(c + 2) * 17 + n], acc);
            acc = fmaf(crow[c + 3], wproj[(c + 3) * 17 + n], acc);
        }
    }
    if (n < 16) {
        const float delta = softplusf(acc + dt_b[n]);
        dAbuf[(size_t)row * 16 + n] = expf(-delta * softplusf(Avec[n]));
    } else if (n == 16) {
        const float g = acc + xproj_b[20];
        gsilu[row] = g / (1.0f + expf(-g));   // silu(g)
    }
}

// =====================================================================
// Selective scan. dA is channel-independent -> stage 128-step chunks of
// dA (8KB) + silu(gamma) in LDS, shared by all 256 channels of the block.
// Thread owns (b,d): h[16] in registers, 32 FMA per step, coalesced x
// loads, in-place epilogue  res <- y*silu(gamma) + res.
// =====================================================================
#define TCHUNK 128

__global__ __launch_bounds__(256)
void k_scan(const float* __restrict__ xcopy, float* __restrict__ resy,
            const float* __restrict__ dAbuf, const float* __restrict__ gsilu,
            const float* __restrict__ Bm, const float* __restrict__ Cm)
{
    __shared__ float sdA[TCHUNK * 16];
    __shared__ float sg[TCHUNK];

    const int tid = threadIdx.x;
    const int d   = blockIdx.x * 256 + tid;
    const int b   = blockIdx.y;

    float h[16], dBv[16], dCv[16];
#pragma unroll
    for (int n = 0; n < 16; ++n) {
        h[n]   = 0.0f;
        dBv[n] = softplusf(Bm[(size_t)d * 16 + n]);
        dCv[n] = Cm[(size_t)d * 16 + n];
    }

    const size_t rowbase = (size_t)b * SSZ;
    for (int t0 = 0; t0 < SSZ; t0 += TCHUNK) {
        __syncthreads();
        {   // cooperative load: TCHUNK*16 = 2048 contiguous floats (8/thread)
            const float* src = dAbuf + (rowbase + t0) * 16;
            *(v4f*)(sdA + tid * 8)     = *(const v4f*)(src + tid * 8);
            *(v4f*)(sdA + tid * 8 + 4) = *(const v4f*)(src + tid * 8 + 4);
            if (tid < TCHUNK) sg[tid] = gsilu[rowbase + t0 + tid];
        }
        __syncthreads();

        for (int tt = 0; tt < TCHUNK; ++tt) {
            const size_t idx = (rowbase + t0 + tt) * DINNER + d;
            const float xv = xcopy[idx];
            float y = 0.0f;
#pragma unroll
            for (int n = 0; n < 16; ++n) {
                h[n] = fmaf(h[n], sdA[tt * 16 + n], xv * dBv[n]);
                y = fmaf(h[n], dCv[n], y);
            }
            resy[idx] = fmaf(y, sg[tt], resy[idx]);   // y*silu(gamma) + res
        }
    }
}

// =====================================================================
extern "C" void kernel_launch(void* const* d_in, const int* in_sizes, int n_in,
                              void* d_out, int out_size, void* d_ws, size_t ws_size,
                              hipStream_t stream) {
    const float* x       = (const float*)d_in[0];   // (B,S,DIM)
    const float* in_w    = (const float*)d_in[1];   // (4096,1024)
    const float* in_b    = (const float*)d_in[2];   // (4096)
    const float* conv_w  = (const float*)d_in[3];   // (2048,1,4)
    const float* conv_b  = (const float*)d_in[4];   // (2048)
    const float* xproj_w = (const float*)d_in[5];   // (21,2048)
    const float* xproj_b = (const float*)d_in[6];   // (21)
    const float* dt_w    = (const float*)d_in[7];   // (16,2048)
    const float* dt_b    = (const float*)d_in[8];   // (16)
    const float* Avec    = (const float*)d_in[9];   // (16)
    const float* Bm      = (const float*)d_in[10];  // (2048,16)
    const float* Cm      = (const float*)d_in[11];  // (2048,16)
    const float* out_w   = (const float*)d_in[12];  // (1024,2048)
    const float* out_b   = (const float*)d_in[13];  // (1024)
    float* out = (float*)d_out;

    float* ws    = (float*)d_ws;
    float* xcopy = ws;                                   // M*DINNER
    float* resy  = xcopy + (size_t)MROWS * DINNER;       // M*DINNER (in-place -> ssm out)
    float* convs = resy  + (size_t)MROWS * DINNER;       // M*DINNER
    float* dAbuf = convs + (size_t)MROWS * DINNER;       // M*16
    float* gsilu = dAbuf + (size_t)MROWS * 16;           // M
    float* wproj = gsilu + MROWS;                        // 2048*17

    // 1) input projection: x(8192x1024) @ in_w^T(1024x4096) + b, split
    k_gemm_f32<<<dim3(4096 / 128, MROWS / 64), 256, GEMM_LDS_BYTES, stream>>>(
        x, in_w, in_b, xcopy, resy, DIM, DINNER, DINNER, DINNER);

    // 2) pack live projection rows (dt 16 + gamma 1), transposed
    k_prep<<<(DINNER * 17 + 255) / 256, 256, 0, stream>>>(dt_w, xproj_w, wproj);

    // 3) depthwise conv along sequence
    k_conv<<<dim3(DINNER / 256, SSZ, BSZ), 256, 0, stream>>>(xcopy, conv_w, conv_b, convs);

    // 4) delta -> dA, gamma -> silu(gamma)
    k_delta<<<MROWS, 32, 0, stream>>>(convs, wproj, dt_b, xproj_b, Avec, dAbuf, gsilu);

    // 5) selective scan + gated residual (in place into resy)
    k_scan<<<dim3(DINNER / 256, BSZ), 256, 0, stream>>>(xcopy, resy, dAbuf, gsilu, Bm, Cm);

    // 6) output projection: resy(8192x2048) @ out_w^T(2048x1024) + b
    k_gemm_f32<<<dim3(DIM / 128, MROWS / 64), 256, GEMM_LDS_BYTES, stream>>>(
        resy, out_w, out_b, out, out, DINNER, 1 << 30, DIM, DIM);
}